// MentionScore_61718680043993
// MI455X (gfx1250) — compile-verified
//
#include <hip/hip_runtime.h>
#include <hip/hip_bf16.h>

// ---------------- problem constants ----------------
#define T_TOK   2048
#define D_DIM   1024
#define L_SPAN  10
#define WD_DIM  20
#define H_DIM   1000     // MLP hidden
#define G_DIM   3092     // 3*D + WD
#define N_SPANS 20435
#define K_TOP   819
#define H_PAD   1024     // padded column count (multiple of 128)
#define KC_A    32       // K-chunks for alpha GEMM (1024/32)
#define KC_G    97       // K-chunks for mention GEMM (3104/32)

#define A_STRIDE 36      // LDS A row stride (floats): 16B aligned, bank-clean
#define B_STRIDE 288     // LDS B pair-row stride (floats): 16B aligned, mod64==32

typedef float v2f __attribute__((ext_vector_type(2)));
typedef float v8f __attribute__((ext_vector_type(8)));

// =====================================================================
// Fused 2-layer row-score MLP:  out[row] = leaky(X @ W1 + b1) @ W2 + b2
// MODE 0: X = states rows (alpha).  MODE 1: X = virtual span-rep g rows.
// Block: 256 threads = 8 waves; BM=64 rows, col-chunks of 128 looped
// in-block; BK=32, double-buffered LDS; WMMA f32 16x16x4.
// B tile is stored K-pair-interleaved so each B fragment is one aligned
// ds_load_b64 landing directly in the VGPR pair WMMA consumes.
// =====================================================================
template <int MODE>
__global__ void __launch_bounds__(256)
mlp_rowscore_kernel(const float* __restrict__ X,
                    const int*   __restrict__ starts,
                    const int*   __restrict__ widths,
                    const float* __restrict__ attn,
                    const float* __restrict__ wtab,
                    const float* __restrict__ W1,
                    const float* __restrict__ B1,
                    const float* __restrict__ W2,
                    const float* __restrict__ B2,
                    float*       __restrict__ outv,
                    int Kdim, int nkc, int nvalid)
{
    __shared__ __align__(16) float As[2][64 * A_STRIDE];   // row-major, 64 rows
    __shared__ __align__(16) float Bs[2][16 * B_STRIDE];   // 16 k-pairs x 128 cols x2
    __shared__ float rowsum[2][64];
    __shared__ int   sS[64], sE[64], sW[64];

    const int tid   = threadIdx.x;
    const int lane  = tid & 31;
    const int wv    = tid >> 5;
    const int wrow  = wv & 3;        // 4 row-strips of 16
    const int wcol  = wv >> 2;       // 2 col-strips of 64
    const int hi    = lane >> 4;     // half-wave select
    const int m     = lane & 15;
    const int rbase = blockIdx.x * 64;

    if (MODE == 1) {
        if (tid < 64) {
            int rg = rbase + tid;
            int s = 0, w = 0;
            if (rg < nvalid) { s = starts[rg]; w = widths[rg]; }
            sS[tid] = s; sW[tid] = w; sE[tid] = s + w;
        }
        __syncthreads();
    }

    auto loadA = [&](int buf, int kc) {
        const int kbase = kc * 32;
#pragma unroll
        for (int rr = 0; rr < 2; ++rr) {
            int fidx = tid + rr * 256;            // [0,512) float4 units
            int row  = fidx >> 3;
            int k    = (fidx & 7) * 4;
            int col  = kbase + k;
            float4 v; v.x = v.y = v.z = v.w = 0.f;
            if (MODE == 0) {
                v = *(const float4*)(X + (size_t)(rbase + row) * D_DIM + col);
            } else {
                int rg = rbase + row;
                if (rg < nvalid) {
                    if (col < 1024) {
                        v = *(const float4*)(X + (size_t)sS[row] * D_DIM + col);
                    } else if (col < 2048) {
                        v = *(const float4*)(X + (size_t)sE[row] * D_DIM + (col - 1024));
                    } else if (col < 3072) {
                        v = *(const float4*)(attn + (size_t)rg * D_DIM + (col - 2048));
                    } else {
                        int c2 = col - 3072;
                        int w  = sW[row];
                        if (c2 + 0 < WD_DIM) v.x = wtab[w * WD_DIM + c2 + 0];
                        if (c2 + 1 < WD_DIM) v.y = wtab[w * WD_DIM + c2 + 1];
                        if (c2 + 2 < WD_DIM) v.z = wtab[w * WD_DIM + c2 + 2];
                        if (c2 + 3 < WD_DIM) v.w = wtab[w * WD_DIM + c2 + 3];
                    }
                }
            }
            *(float4*)&As[buf][row * A_STRIDE + k] = v;
        }
    };

    // B tile loader: reads two consecutive W1 K-rows as float2 each and
    // stores one aligned b128 into the pair-interleaved layout:
    //   Bs[k2*B_STRIDE + 2c .. 2c+3] = {E[c], O[c], E[c+1], O[c+1]}
    auto loadB = [&](int buf, int kc, int cb) {
        const int kbase = kc * 32;
#pragma unroll
        for (int rr = 0; rr < 4; ++rr) {
            int fidx = tid + rr * 256;            // [0,1024): (k2, col-pair)
            int k2   = fidx >> 6;                 // 16 k-pairs
            int c    = (fidx & 63) * 2;           // even column
            int colg = cb + c;
            int kg0  = kbase + 2 * k2;
            int kg1  = kg0 + 1;
            float2 e; e.x = e.y = 0.f;
            float2 o; o.x = o.y = 0.f;
            if (colg < H_DIM) {                   // colg even, H_DIM even
                if (kg0 < Kdim) {
                    const float* p = W1 + (size_t)kg0 * H_DIM + colg;
                    e = *(const float2*)p;
                    if (rr == 0 && kg0 + 32 < Kdim)
                        __builtin_prefetch(p + (size_t)32 * H_DIM, 0, 0);
                }
                if (kg1 < Kdim) {
                    o = *(const float2*)(W1 + (size_t)kg1 * H_DIM + colg);
                }
            }
            float4 v; v.x = e.x; v.y = o.x; v.z = e.y; v.w = o.y;
            *(float4*)&Bs[buf][k2 * B_STRIDE + 2 * c] = v;
        }
    };

    float prt[8];
#pragma unroll
    for (int r = 0; r < 8; ++r) prt[r] = 0.f;

    v8f acc[4];

    for (int cb = 0; cb < H_PAD; cb += 128) {
        float w2v[4], b1v[4];
#pragma unroll
        for (int t = 0; t < 4; ++t) {
            int colL = cb + wcol * 64 + t * 16 + m;
            w2v[t] = (colL < H_DIM) ? W2[colL] : 0.f;
            b1v[t] = (colL < H_DIM) ? B1[colL] : 0.f;
#pragma unroll
            for (int e = 0; e < 8; ++e) acc[t][e] = 0.f;
        }

        loadA(0, 0); loadB(0, 0, cb);
        __syncthreads();

        for (int kc = 0; kc < nkc; ++kc) {
            const int cur = kc & 1;
            if (kc + 1 < nkc) { loadA(cur ^ 1, kc + 1); loadB(cur ^ 1, kc + 1, cb); }

            const float* Al = As[cur];
            const float* Bl = Bs[cur];
#pragma unroll
            for (int ks = 0; ks < 32; ks += 4) {
                // A fragment: {A[m][ks+2hi], A[m][ks+2hi+1]} — contiguous b64
                v2f af = *(const v2f*)&Al[m * A_STRIDE + ks + 2 * hi];
#pragma unroll
                for (int t = 0; t < 4; ++t) {
                    const int cc = wcol * 64 + t * 16 + m;
                    // B fragment: rows (ks+2hi, ks+2hi+1) at col cc — one b64
                    v2f bf = *(const v2f*)&Bl[(ks / 2 + hi) * B_STRIDE + 2 * cc];
                    acc[t] = __builtin_amdgcn_wmma_f32_16x16x4_f32(
                        false, af, false, bf, (short)0, acc[t], false, false);
                }
            }
            __syncthreads();
        }

        // second layer: leaky-relu + contract columns into per-row partials
#pragma unroll
        for (int t = 0; t < 4; ++t) {
#pragma unroll
            for (int r = 0; r < 8; ++r) {
                float h = acc[t][r] + b1v[t];
                h = (h > 0.f) ? h : 0.01f * h;
                prt[r] += h * w2v[t];
            }
        }
    }

    // reduce within each 16-lane half (C/D layout: row = r + 8*hi, col = lane&15)
#pragma unroll
    for (int r = 0; r < 8; ++r) {
        float v = prt[r];
        v += __shfl_xor(v, 1, 32);
        v += __shfl_xor(v, 2, 32);
        v += __shfl_xor(v, 4, 32);
        v += __shfl_xor(v, 8, 32);
        prt[r] = v;
    }
    if (lane == 0) {
#pragma unroll
        for (int r = 0; r < 8; ++r) rowsum[wcol][wrow * 16 + r] = prt[r];
    }
    if (lane == 16) {
#pragma unroll
        for (int r = 0; r < 8; ++r) rowsum[wcol][wrow * 16 + 8 + r] = prt[r];
    }
    __syncthreads();
    if (tid < 64) {
        int rg = rbase + tid;
        float s = rowsum[0][tid] + rowsum[1][tid] + B2[0];
        if (MODE == 0) outv[rg] = s;
        else if (rg < nvalid) outv[rg] = s;
    }
}

// =====================================================================
// Per-span softmax attention over <=10 tokens -> attn_emb [N, D]
// =====================================================================
__global__ void __launch_bounds__(256)
span_attn_kernel(const float* __restrict__ states,
                 const int*   __restrict__ starts,
                 const int*   __restrict__ widths,
                 const float* __restrict__ alpha,
                 float*       __restrict__ attn)
{
    const int n = blockIdx.x;
    const int s = starts[n];
    const int w = widths[n];
    const int tid = threadIdx.x;

    __shared__ float lg[L_SPAN];
    __shared__ float wts[L_SPAN];

    if (tid < L_SPAN) {
        int p = s + tid;
        bool valid = (tid <= w) && (p < T_TOK);
        lg[tid] = valid ? alpha[p < T_TOK ? p : (T_TOK - 1)] : -1e9f;
    }
    __syncthreads();
    if (tid < L_SPAN) {
        float mx = -1e30f;
        for (int j = 0; j < L_SPAN; ++j) mx = fmaxf(mx, lg[j]);
        float sum = 0.f;
        for (int j = 0; j < L_SPAN; ++j) sum += __expf(lg[j] - mx);
        wts[tid] = __expf(lg[tid] - mx) / sum;
    }
    __syncthreads();

    const int d = tid * 4;
    float a0 = 0.f, a1 = 0.f, a2 = 0.f, a3 = 0.f;
#pragma unroll
    for (int j = 0; j < L_SPAN; ++j) {
        float wj = wts[j];
        int p = s + j; p = (p < T_TOK - 1) ? p : (T_TOK - 1);
        const float4 v = *(const float4*)(states + (size_t)p * D_DIM + d);
        a0 += wj * v.x; a1 += wj * v.y; a2 += wj * v.z; a3 += wj * v.w;
    }
    float4 o; o.x = a0; o.y = a1; o.z = a2; o.w = a3;
    *(float4*)(attn + (size_t)n * D_DIM + d) = o;
}

// =====================================================================
// Exact deterministic top-K via 4-pass radix-select + bitonic sort
// =====================================================================
struct Ctl { unsigned prefix, remaining, c_gt, c_eq; };

__device__ __forceinline__ unsigned fkey(float f) {
    unsigned u = __float_as_uint(f);
    return (u & 0x80000000u) ? ~u : (u | 0x80000000u);
}

__global__ void topk_init_kernel(unsigned* hist, Ctl* ctl)
{
    hist[threadIdx.x] = 0u;
    if (threadIdx.x == 0) {
        ctl->prefix = 0u; ctl->remaining = K_TOP; ctl->c_gt = 0u; ctl->c_eq = 0u;
    }
}

__global__ void __launch_bounds__(256)
topk_hist_kernel(const float* __restrict__ scores, unsigned* __restrict__ hist,
                 const Ctl* __restrict__ ctl, int pass, int n)
{
    __shared__ unsigned h[256];
    const int tid = threadIdx.x;
    h[tid] = 0u;
    __syncthreads();
    int i = blockIdx.x * 256 + tid;
    if (i < n) {
        unsigned u = fkey(scores[i]);
        bool match = (pass == 3) || ((u >> ((pass + 1) * 8)) == ctl->prefix);
        if (match) atomicAdd(&h[(u >> (pass * 8)) & 255u], 1u);
    }
    __syncthreads();
    if (h[tid]) atomicAdd(&hist[tid], h[tid]);
}

__global__ void topk_scan_kernel(unsigned* hist, Ctl* ctl,
                                 unsigned long long* sel, int pass)
{
    const int tid = threadIdx.x;
    if (tid == 0) {
        unsigned rem = ctl->remaining;
        unsigned cum = 0u;
        int bsel = 0;
        for (int b = 255; b >= 0; --b) {
            unsigned c = hist[b];
            if (cum + c >= rem) { bsel = b; break; }
            cum += c;
        }
        ctl->prefix    = (pass == 3) ? (unsigned)bsel : ((ctl->prefix << 8) | (unsigned)bsel);
        ctl->remaining = rem - cum;
        if (pass == 0) { ctl->c_gt = 0u; ctl->c_eq = 0u; }
    }
    __syncthreads();
    hist[tid] = 0u;
    if (pass == 0) {
        for (int i = tid; i < 1024; i += 256) sel[i] = 0ull;
    }
}

__global__ void __launch_bounds__(256)
topk_select_kernel(const float* __restrict__ scores, Ctl* ctl,
                   unsigned long long* __restrict__ sel, int n)
{
    int i = blockIdx.x * 256 + threadIdx.x;
    if (i >= n) return;
    unsigned u = fkey(scores[i]);
    unsigned thresh = ctl->prefix;
    unsigned long long comp =
        (((unsigned long long)u) << 32) | (unsigned long long)(0xFFFFFFFFu - (unsigned)i);
    if (u > thresh) {
        unsigned pos = atomicAdd(&ctl->c_gt, 1u);
        sel[pos] = comp;
    } else if (u == thresh) {
        unsigned t = atomicAdd(&ctl->c_eq, 1u);
        if (t < ctl->remaining) sel[(K_TOP - ctl->remaining) + t] = comp;
    }
}

__global__ void __launch_bounds__(512)
topk_sort_kernel(unsigned long long* sel)
{
    __shared__ unsigned long long key[1024];
    const int tid = threadIdx.x;
    key[tid]       = sel[tid];
    key[tid + 512] = sel[tid + 512];
    __syncthreads();
    for (int kk = 2; kk <= 1024; kk <<= 1) {
        for (int j = kk >> 1; j > 0; j >>= 1) {
            for (int idx = tid; idx < 1024; idx += 512) {
                int p = idx ^ j;
                if (p > idx) {
                    bool desc = ((idx & kk) == 0);
                    unsigned long long a = key[idx], b = key[p];
                    if ((a < b) == desc) { key[idx] = b; key[p] = a; }
                }
            }
            __syncthreads();
        }
    }
    sel[tid]       = key[tid];
    sel[tid + 512] = key[tid + 512];
}

__global__ void __launch_bounds__(256)
topk_gather_kernel(const unsigned long long* __restrict__ sel,
                   const float* __restrict__ scores,
                   const float* __restrict__ states,
                   const int*   __restrict__ starts,
                   const int*   __restrict__ widths,
                   const float* __restrict__ attn,
                   const float* __restrict__ wtab,
                   float*       __restrict__ out)
{
    const int k = blockIdx.x;
    unsigned long long comp = sel[k];
    const int n = (int)(0xFFFFFFFFu - (unsigned)(comp & 0xFFFFFFFFull));
    const int s = starts[n];
    const int w = widths[n];
    const int e = s + w;
    const int tid = threadIdx.x;
    if (tid == 0) out[k] = scores[n];
    float* g = out + K_TOP + (size_t)k * G_DIM;
    for (int c = tid; c < G_DIM; c += 256) {
        float v;
        if      (c < 1024) v = states[(size_t)s * D_DIM + c];
        else if (c < 2048) v = states[(size_t)e * D_DIM + (c - 1024)];
        else if (c < 3072) v = attn[(size_t)n * D_DIM + (c - 2048)];
        else               v = wtab[w * WD_DIM + (c - 3072)];
        g[c] = v;
    }
}

// =====================================================================
// Host-side launch
// =====================================================================
static inline size_t alignUp(size_t x, size_t a) { return (x + a - 1) / a * a; }

extern "C" void kernel_launch(void* const* d_in, const int* in_sizes, int n_in,
                              void* d_out, int out_size, void* d_ws, size_t ws_size,
                              hipStream_t stream)
{
    (void)in_sizes; (void)n_in; (void)out_size; (void)ws_size;

    const float* states = (const float*)d_in[0];
    const int*   starts = (const int*)  d_in[1];
    const int*   widths = (const int*)  d_in[2];
    const float* wtab   = (const float*)d_in[3];
    const float* Wa1    = (const float*)d_in[4];
    const float* ba1    = (const float*)d_in[5];
    const float* Wa2    = (const float*)d_in[6];
    const float* ba2    = (const float*)d_in[7];
    const float* Wm1    = (const float*)d_in[8];
    const float* bm1    = (const float*)d_in[9];
    const float* Wm2    = (const float*)d_in[10];
    const float* bm2    = (const float*)d_in[11];

    char* ws = (char*)d_ws;
    size_t off = 0;
    float* attn   = (float*)(ws + off); off = alignUp(off + (size_t)N_SPANS * D_DIM * 4, 256);
    float* alpha  = (float*)(ws + off); off = alignUp(off + (size_t)T_TOK * 4, 256);
    float* scores = (float*)(ws + off); off = alignUp(off + (size_t)N_SPANS * 4, 256);
    unsigned* hist = (unsigned*)(ws + off); off = alignUp(off + 256 * 4, 256);
    Ctl* ctl      = (Ctl*)(ws + off);   off = alignUp(off + sizeof(Ctl), 256);
    unsigned long long* sel = (unsigned long long*)(ws + off); off += 1024 * 8;

    float* out = (float*)d_out;

    // 1) alpha = leaky(states@Wa1+ba1)@Wa2+ba2          (WMMA f32)
    mlp_rowscore_kernel<0><<<T_TOK / 64, 256, 0, stream>>>(
        states, nullptr, nullptr, nullptr, nullptr,
        Wa1, ba1, Wa2, ba2, alpha, D_DIM, KC_A, T_TOK);

    // 2) span attention embeddings
    span_attn_kernel<<<N_SPANS, 256, 0, stream>>>(states, starts, widths, alpha, attn);

    // 3) mention scores = leaky(g@Wm1+bm1)@Wm2+bm2      (WMMA f32, g built on the fly)
    mlp_rowscore_kernel<1><<<(N_SPANS + 63) / 64, 256, 0, stream>>>(
        states, starts, widths, attn, wtab,
        Wm1, bm1, Wm2, bm2, scores, G_DIM, KC_G, N_SPANS);

    // 4) exact top-K (radix select, 4 byte passes) + canonical sort + gather
    topk_init_kernel<<<1, 256, 0, stream>>>(hist, ctl);
    const int nblk = (N_SPANS + 255) / 256;
    for (int p = 3; p >= 0; --p) {
        topk_hist_kernel<<<nblk, 256, 0, stream>>>(scores, hist, ctl, p, N_SPANS);
        topk_scan_kernel<<<1, 256, 0, stream>>>(hist, ctl, sel, p);
    }
    topk_select_kernel<<<nblk, 256, 0, stream>>>(scores, ctl, sel, N_SPANS);
    topk_sort_kernel<<<1, 512, 0, stream>>>(sel);
    topk_gather_kernel<<<K_TOP, 256, 0, stream>>>(sel, scores, states, starts,
                                                  widths, attn, wtab, out);
}